// rdf_40338332844610
// MI455X (gfx1250) — compile-verified
//
#include <hip/hip_runtime.h>
#include <hip/hip_bf16.h>
#include <math.h>

// ---------------- problem constants (from reference) ----------------
#define NATOMS  600
#define NBATCH  4
#define NBINS   64
#define NPAIRB  (NATOMS * NATOMS)          // 360000 pairs per batch
#define NPAIRS  (NBATCH * NPAIRB)          // 1440000 total pair slots

#define BOX_L   20.0f
#define BOX_HL  10.0f
#define BINW    (7.5f / 64.0f)             // histogram bin width (BINS spacing)
#define GWIDTH  (7.5f / 63.0f)             // Gaussian offset spacing
#define GCOEFF  (-0.5f / (GWIDTH * GWIDTH))// = -35.28
#define CUT_ADJ (7.5f + 2.0f * BINW)       // 7.734375
#define CUT2    (CUT_ADJ * CUT_ADJ)

typedef __attribute__((ext_vector_type(2))) float v2f;
typedef __attribute__((ext_vector_type(8))) float v8f;

// ---------------- kernel 1: zero the 64-bin global accumulator ----------------
__global__ void rdf_zero_kernel(float* __restrict__ count) {
    if (threadIdx.x < NBINS) count[threadIdx.x] = 0.0f;
}

// ---------------- kernel 2: pair loop + Gaussian smear + WMMA accumulate ------
// Wave-level scheme: each lane owns one pair (distance) per outer iteration;
// 8 sub-steps consume the 32 distances in quads of 4. Per quad, the wave
// issues 4x v_wmma_f32_16x16x4_f32 (one per 16-bin group) with an all-ones
// A matrix, so C[m,n] += sum_k B[k,n] accumulates the soft histogram on the
// matrix pipe while the VALU/TRANS pipes run v_exp_f32.
__global__ __launch_bounds__(256) void rdf_pairs_kernel(
        const float* __restrict__ xyz, float* __restrict__ count) {
    __shared__ float sxyz[NBATCH * NATOMS * 3];   // 28.8 KB, whole input
    for (int idx = threadIdx.x; idx < NBATCH * NATOMS * 3; idx += blockDim.x)
        sxyz[idx] = xyz[idx];
    __syncthreads();

    const int lane   = threadIdx.x & 31;
    const int wave   = blockIdx.x * (blockDim.x >> 5) + (threadIdx.x >> 5);
    const int nwaves = gridDim.x * (blockDim.x >> 5);
    const int nIter  = (NPAIRS + nwaves * 32 - 1) / (nwaves * 32);

    // A matrix = all ones (16x4 f32 -> 2 VGPRs, every lane {1,1})
    v2f aOnes; aOnes.x = 1.0f; aOnes.y = 1.0f;
    v8f acc[4] = {};                              // 4 bin-groups of 16 bins

    const int   kOff    = (lane < 16) ? 0 : 2;    // which quad members this half supplies
    const float offLane = (float)(lane & 15) * GWIDTH;

    for (int it = 0; it < nIter; ++it) {
        int  t     = (it * nwaves + wave) * 32 + lane;
        bool valid = (t < NPAIRS);
        int  tc    = valid ? t : 0;
        int  b     = tc / NPAIRB;
        int  r     = tc - b * NPAIRB;
        int  i     = r / NATOMS;
        int  j     = r - i * NATOMS;

        const float* pi = &sxyz[(b * NATOMS + i) * 3];
        const float* pj = &sxyz[(b * NATOMS + j) * 3];
        float dx = pj[0] - pi[0];
        float dy = pj[1] - pi[1];
        float dz = pj[2] - pi[2];
        // minimum-image wrap (matches reference's offset logic)
        dx += (dx < -BOX_HL ? BOX_L : 0.0f) - (dx >= BOX_HL ? BOX_L : 0.0f);
        dy += (dy < -BOX_HL ? BOX_L : 0.0f) - (dy >= BOX_HL ? BOX_L : 0.0f);
        dz += (dz < -BOX_HL ? BOX_L : 0.0f) - (dz >= BOX_HL ? BOX_L : 0.0f);
        float d2 = dx * dx + dy * dy + dz * dz;
        bool  m  = valid && (d2 != 0.0f) && (d2 < CUT2);
        // invalid pairs -> huge distance -> all Gaussians underflow to 0.0
        float d  = m ? sqrtf(d2) : 1.0e9f;

        #pragma unroll
        for (int s = 0; s < 8; ++s) {             // 8 quads of 4 pairs
            float dA = __shfl(d, s * 4 + kOff,     32);
            float dB = __shfl(d, s * 4 + kOff + 1, 32);
            #pragma unroll
            for (int g = 0; g < 4; ++g) {         // 4 groups of 16 bins
                float off = offLane + (float)(16 * g) * GWIDTH;
                float eA  = dA - off;
                float eB  = dB - off;
                v2f bv;
                bv.x = __expf(GCOEFF * eA * eA);
                bv.y = __expf(GCOEFF * eB * eB);
                acc[g] = __builtin_amdgcn_wmma_f32_16x16x4_f32(
                    /*neg_a=*/false, aOnes, /*neg_b=*/false, bv,
                    /*c_mod=*/(short)0, acc[g],
                    /*reuse_a=*/false, /*reuse_b=*/false);
            }
        }
    }

    // All 16 rows of C are identical; lane L<16 VGPR0 = count for bin (16g + L).
    if (lane < 16) {
        #pragma unroll
        for (int g = 0; g < 4; ++g)
            atomicAdd(&count[g * 16 + lane], acc[g][0]);
    }
}

// ---------------- kernel 3: normalize, write BINS + rdf --------------------
__global__ void rdf_finalize_kernel(const float* __restrict__ count,
                                    float* __restrict__ out) {
    __shared__ float sc[NBINS];
    __shared__ float stot;
    int tid = threadIdx.x;                        // blockDim = 128
    if (tid < NBINS) sc[tid] = count[tid];
    __syncthreads();
    if (tid == 0) {
        float tot = 0.0f;
        for (int k = 0; k < NBINS; ++k) tot += sc[k];
        stot = tot;
    }
    __syncthreads();

    if (tid < NBINS + 1)                          // BINS[0..64]
        out[tid] = (float)tid * BINW;
    if (tid < NBINS) {                            // rdf_val[0..63]
        float b0  = (float)tid * BINW;
        float b1  = (float)(tid + 1) * BINW;
        float vol = (4.0f * 3.14159265358979f / 3.0f) * (b1 * b1 * b1 - b0 * b0 * b0);
        float K   = (2.0f * CUT_ADJ) * (2.0f * CUT_ADJ) * (2.0f * CUT_ADJ);
        out[NBINS + 1 + tid] = (sc[tid] / stot) * K / (2.0f * vol);
    }
}

// ---------------- launch ----------------
extern "C" void kernel_launch(void* const* d_in, const int* in_sizes, int n_in,
                              void* d_out, int out_size, void* d_ws, size_t ws_size,
                              hipStream_t stream) {
    const float* xyz   = (const float*)d_in[0];   // [4,600,3] f32
    float*       out   = (float*)d_out;           // 129 f32
    float*       count = (float*)d_ws;            // 64-bin global accumulator
    (void)in_sizes; (void)n_in; (void)out_size; (void)ws_size;

    rdf_zero_kernel<<<1, 64, 0, stream>>>(count);
    rdf_pairs_kernel<<<256, 256, 0, stream>>>(xyz, count);   // 2048 wave32s
    rdf_finalize_kernel<<<1, 128, 0, stream>>>(count, out);
}